// ModelNew_32607391711573
// MI455X (gfx1250) — compile-verified
//
#include <hip/hip_runtime.h>
#include <math.h>

// ---------------------------------------------------------------------------
// Fused: h = x @ W^T + b ; per-row instance-norm over 2048 feats ;
//        out = (norm(h)*g + beta + y) * y
// x: [B,128] f32, W: [2048,128] f32 (row-major), y/out: [B,2048] f32
//
// MI455X: HBM floor ~24us (y in + out out dominate). GEMM via
// V_WMMA_F32_16X16X4_F32 (fp32-exact). W (1MB) stays L2-resident.
// B fragments ping-pong between two register buffers (no rotate copies).
// x tile staged via GLOBAL_LOAD_ASYNC_TO_LDS_B128 (ASYNCcnt path).
// ---------------------------------------------------------------------------

typedef __attribute__((ext_vector_type(2))) float v2f;
typedef __attribute__((ext_vector_type(8))) float v8f;
typedef int v4i __attribute__((vector_size(4 * sizeof(int))));

#define IN_F   128
#define OUT_F  2048
#define EPSV   1e-5f

#define BLOCK_THREADS 512          // 16 waves (wave32)
#define TILES_PER_WAVE 8           // 2048/16 = 128 N-tiles / 16 waves
#define XPAD  130                  // padded x-tile row stride (bank-conflict-free)

#if defined(__has_builtin)
#if __has_builtin(__builtin_amdgcn_global_load_async_to_lds_b128) && \
    __has_builtin(__builtin_amdgcn_s_wait_asynccnt)
#define USE_ASYNC_LDS 1
#endif
#endif
#ifndef USE_ASYNC_LDS
#define USE_ASYNC_LDS 0
#endif

typedef __attribute__((address_space(1))) v4i* glb_v4i_p;
typedef __attribute__((address_space(3))) v4i* lds_v4i_p;

__global__ __launch_bounds__(BLOCK_THREADS)
void fused_linear_instnorm_mul(const float* __restrict__ x,
                               const float* __restrict__ y,
                               const float* __restrict__ W,
                               const float* __restrict__ bias,
                               const float* __restrict__ gw,
                               const float* __restrict__ gb,
                               float* __restrict__ out)
{
    __shared__ float lx[16 * XPAD];   // x tile, padded rows
    __shared__ float redS[16];        // per-row sum
    __shared__ float redQ[16];        // per-row sum of squares
    __shared__ float meanS[16];
    __shared__ float istdS[16];

    const int tid  = threadIdx.x;
    const int wave = tid >> 5;
    const int lane = tid & 31;
    const int half = lane >> 4;       // 0: lanes 0-15, 1: lanes 16-31
    const int l16  = lane & 15;
    const int r0   = blockIdx.x * 16; // this block's 16 rows

    if (tid < 16) { redS[tid] = 0.0f; redQ[tid] = 0.0f; }

    // ---- stage x tile (16x128 f32, 8KB contiguous) into padded LDS --------
#if USE_ASYNC_LDS
    {
        // one 16B chunk per thread: row = tid>>5, col = (tid&31)*4
        const int rr = tid >> 5;
        const int c4 = (tid & 31) << 2;
        glb_v4i_p gsrc = (glb_v4i_p)(x + (size_t)r0 * IN_F + (size_t)tid * 4);
        lds_v4i_p ldst = (lds_v4i_p)(lx + rr * XPAD + c4);
        __builtin_amdgcn_global_load_async_to_lds_b128(gsrc, ldst, 0, 0);
    }
#else
    for (int idx = tid; idx < 16 * IN_F; idx += BLOCK_THREADS) {
        const int rr = idx >> 7;
        const int cc = idx & 127;
        lx[rr * XPAD + cc] = x[(size_t)(r0 + rr) * IN_F + cc];
    }
#endif

    // ---- warm L2 with this block's y tile (128KB) while GEMM runs ---------
    {
        const char* ybase = (const char*)(y + (size_t)r0 * OUT_F);
        __builtin_prefetch(ybase + (size_t)tid * 256,       0, 1);
        __builtin_prefetch(ybase + (size_t)tid * 256 + 128, 0, 1);
    }

#if USE_ASYNC_LDS
    __builtin_amdgcn_s_wait_asynccnt(0);
#endif
    __syncthreads();

    // ---- GEMM: 8 N-tiles per wave, K=128 in steps of 4 --------------------
    // A frag (16x4 f32): lane m=l16; VGPRs = x[m][k0+2*half], x[m][k0+2*half+1]
    // B frag (4x16 f32): lane n=l16; VGPRs = W[n][k0+2*half], W[n][k0+2*half+1]
    v8f acc[TILES_PER_WAVE] = {};
    const int ntile0 = wave * TILES_PER_WAVE;

    // per-tile W row base (includes the half-lane K offset)
    const float* wp[TILES_PER_WAVE];
    #pragma unroll
    for (int j = 0; j < TILES_PER_WAVE; ++j)
        wp[j] = W + (size_t)((ntile0 + j) * 16 + l16) * IN_F + 2 * half;

    // ping-pong B buffers: b0 consumed while b1 loads, then swap roles.
    v2f b0[TILES_PER_WAVE], b1[TILES_PER_WAVE];
    #pragma unroll
    for (int j = 0; j < TILES_PER_WAVE; ++j)
        b0[j] = *(const v2f*)(wp[j]);

    for (int k0 = 0; k0 < IN_F; k0 += 8) {
        const int ka = k0 + 4;                           // always <= 124
        const int kb = (k0 + 8 < IN_F) ? (k0 + 8) : 0;   // wrap on last iter

        const v2f a0 = *(const v2f*)&lx[l16 * XPAD + k0 + 2 * half];
        #pragma unroll
        for (int j = 0; j < TILES_PER_WAVE; ++j)
            b1[j] = *(const v2f*)(wp[j] + ka);
        #pragma unroll
        for (int j = 0; j < TILES_PER_WAVE; ++j)
            acc[j] = __builtin_amdgcn_wmma_f32_16x16x4_f32(
                false, a0, false, b0[j], (short)0, acc[j], false, false);

        const v2f a1 = *(const v2f*)&lx[l16 * XPAD + ka + 2 * half];
        #pragma unroll
        for (int j = 0; j < TILES_PER_WAVE; ++j)
            b0[j] = *(const v2f*)(wp[j] + kb);
        #pragma unroll
        for (int j = 0; j < TILES_PER_WAVE; ++j)
            acc[j] = __builtin_amdgcn_wmma_f32_16x16x4_f32(
                false, a1, false, b1[j], (short)0, acc[j], false, false);
    }

    // ---- bias add + per-lane partial row sums -----------------------------
    // C/D layout: VGPR v, lane l -> (M = v + 8*half, N = l16)
    float ps[8], pq[8];
    #pragma unroll
    for (int v = 0; v < 8; ++v) { ps[v] = 0.0f; pq[v] = 0.0f; }

    #pragma unroll
    for (int j = 0; j < TILES_PER_WAVE; ++j) {
        const float bj = bias[(ntile0 + j) * 16 + l16];
        #pragma unroll
        for (int v = 0; v < 8; ++v) {
            const float h = acc[j][v] + bj;
            acc[j][v] = h;
            ps[v] += h;
            pq[v] += h * h;
        }
    }

    // butterfly-reduce across the 16 lanes of each half (wave32 shuffles)
    #pragma unroll
    for (int mask = 8; mask >= 1; mask >>= 1) {
        #pragma unroll
        for (int v = 0; v < 8; ++v) {
            ps[v] += __shfl_xor(ps[v], mask, 32);
            pq[v] += __shfl_xor(pq[v], mask, 32);
        }
    }
    // lane (l16==v) publishes row v + 8*half via LDS float atomics
    if (l16 < 8) {
        const int v = l16;
        atomicAdd(&redS[v + 8 * half], ps[v]);
        atomicAdd(&redQ[v + 8 * half], pq[v]);
    }
    __syncthreads();

    if (tid < 16) {
        const float m = redS[tid] * (1.0f / OUT_F);
        const float q = redQ[tid] * (1.0f / OUT_F);
        meanS[tid] = m;
        istdS[tid] = rsqrtf(q - m * m + EPSV);
    }
    __syncthreads();

    float mrow[8], irow[8];
    #pragma unroll
    for (int v = 0; v < 8; ++v) {
        mrow[v] = meanS[v + 8 * half];
        irow[v] = istdS[v + 8 * half];
    }

    // ---- epilogue: normalize, fuse with y, NT-store the output ------------
    #pragma unroll
    for (int j = 0; j < TILES_PER_WAVE; ++j) {
        const int   n  = (ntile0 + j) * 16 + l16;
        const float wn = gw[n];
        const float bn = gb[n];
        #pragma unroll
        for (int v = 0; v < 8; ++v) {
            const int    row = r0 + v + 8 * half;
            const size_t off = (size_t)row * OUT_F + n;
            const float  yv  = y[off];
            const float  nv  = (acc[j][v] - mrow[v]) * irow[v] * wn + bn;
            __builtin_nontemporal_store((nv + yv) * yv, out + off);
        }
    }
}

extern "C" void kernel_launch(void* const* d_in, const int* in_sizes, int n_in,
                              void* d_out, int out_size, void* d_ws, size_t ws_size,
                              hipStream_t stream) {
    const float* x  = (const float*)d_in[0];   // [B,128]
    const float* y  = (const float*)d_in[1];   // [B,2048]
    const float* W  = (const float*)d_in[2];   // [2048,128]
    const float* lb = (const float*)d_in[3];   // [2048]
    const float* nw = (const float*)d_in[4];   // [2048]
    const float* nb = (const float*)d_in[5];   // [2048]
    float* out = (float*)d_out;

    const int Brows = in_sizes[0] / IN_F;      // 32768
    dim3 grid(Brows / 16);
    dim3 block(BLOCK_THREADS);
    hipLaunchKernelGGL(fused_linear_instnorm_mul, grid, block, 0, stream,
                       x, y, W, lb, nw, nb, out);
}